// ModelDynamic_11570641895673
// MI455X (gfx1250) — compile-verified
//
#include <hip/hip_runtime.h>
#include <float.h>

// ---------------------------------------------------------------------------
// Kernel 1: per-row fill + ancestor patch.
//   grid = n blocks (one per row r), 256 threads.
//   Bulk fill via CDNA5 async LDS->global B128 stores (ASYNCcnt path),
//   then patch <=19 ancestor columns with mask=1.0 / masked=scores[r][a].
// ---------------------------------------------------------------------------
__global__ void rowfill_kernel(const int* __restrict__ mask_index,
                               const float* __restrict__ scores,
                               float* __restrict__ out_mask,
                               float* __restrict__ out_masked,
                               int n, int max_depth) {
  const int r   = blockIdx.x;
  const int tid = threadIdx.x;

  __shared__ int   chain[64];
  __shared__ int   clen;
  __shared__ float s_zero[256 * 4];
  __shared__ float s_neg[256 * 4];

  if (tid == 0) {
    // Ancestor chain: r, p(r), ..., capped at max_depth+1 steps (matches the
    // max_depth+1 OR-propagation iterations in _build_tree), plus column 0.
    int len = 0;
    int a = r;
    chain[len++] = a;
    for (int s = 0; s < max_depth + 1 && a > 0; ++s) {
      a = mask_index[a - 1];
      chain[len++] = a;
    }
    if (chain[len - 1] != 0) chain[len++] = 0;
    clen = len;
  }

  // Constant staging: 16 bytes per lane for each fill value.
#pragma unroll
  for (int k = 0; k < 4; ++k) {
    s_zero[tid * 4 + k] = 0.0f;
    s_neg[tid * 4 + k]  = -FLT_MAX;   // == jnp.finfo(float32).min
  }
  __syncthreads();

  float* mask_row = out_mask   + (size_t)r * n;
  float* msk_row  = out_masked + (size_t)r * n;

  // addrspace(3) offset survives in the low 32 bits of the generic pointer.
  unsigned lds_zero = (unsigned)(size_t)(&s_zero[tid * 4]);
  unsigned lds_neg  = (unsigned)(size_t)(&s_neg[tid * 4]);

  const int nchunks = n >> 2;  // float4 chunks per row
  for (int c4 = tid; c4 < nchunks; c4 += blockDim.x) {
    unsigned long long a0 = (unsigned long long)(mask_row + (size_t)c4 * 4);
    unsigned long long a1 = (unsigned long long)(msk_row  + (size_t)c4 * 4);
    asm volatile("global_store_async_from_lds_b128 %0, %1, off"
                 :: "v"(a0), "v"(lds_zero) : "memory");
    asm volatile("global_store_async_from_lds_b128 %0, %1, off"
                 :: "v"(a1), "v"(lds_neg)  : "memory");
  }
  // Drain async stores so the ancestor patches below land after the fill.
  asm volatile("s_wait_asynccnt 0" ::: "memory");
  __syncthreads();

  if (tid < clen) {
    int a = chain[tid];
    mask_row[a] = 1.0f;
    msk_row[a]  = scores[(size_t)r * n + a];  // only gather scores we keep
  }
}

// ---------------------------------------------------------------------------
// retrieve_indices pipeline (tiny): flags -> leaf compaction -> walk.
// ---------------------------------------------------------------------------
__global__ void zero_flags_kernel(int* flags, int n) {
  int i = blockIdx.x * blockDim.x + threadIdx.x;
  if (i < n) flags[i] = 0;
}

__global__ void mark_parents_kernel(const int* __restrict__ mask_index,
                                    int* __restrict__ flags, int T, int n) {
  int j = blockIdx.x * blockDim.x + threadIdx.x;
  if (j < T) {
    int v = mask_index[j];
    if (v >= 0 && v < n) flags[v] = 1;  // races all write 1: benign
  }
}

// Single-block stable compaction of indices where flags==0; pad with -1.
__global__ void leaf_scan_kernel(const int* __restrict__ flags,
                                 int* __restrict__ leaf, int n) {
  __shared__ int partial[1024];
  const int t   = threadIdx.x;
  const int nt  = blockDim.x;
  const int per = (n + nt - 1) / nt;
  const int lo  = t * per;
  const int hi  = min(n, lo + per);

  int cnt = 0;
  for (int k = lo; k < hi; ++k) cnt += (flags[k] == 0);
  partial[t] = cnt;
  __syncthreads();

  // Hillis-Steele inclusive scan over nt partials.
  for (int off = 1; off < nt; off <<= 1) {
    int v   = partial[t];
    int add = (t >= off) ? partial[t - off] : 0;
    __syncthreads();
    partial[t] = v + add;
    __syncthreads();
  }
  int total = partial[nt - 1];
  int pos   = partial[t] - cnt;  // exclusive prefix

  for (int k = lo; k < hi; ++k)
    if (flags[k] == 0) leaf[pos++] = k;

  for (int k = total + t; k < n; k += nt) leaf[k] = -1;
}

__global__ void retrieve_kernel(const int* __restrict__ leaf,
                                const int* __restrict__ mask_index,
                                const int* __restrict__ tpid,
                                float* __restrict__ out_ri,
                                int n, int max_depth) {
  int i = blockIdx.x * blockDim.x + threadIdx.x;
  if (i >= n) return;
  int cur = leaf[i];
  float* row = out_ri + (size_t)i * max_depth;
  for (int d = 0; d < max_depth; ++d) {
    int  cc    = (cur < 0) ? 0 : cur;
    int  pos   = tpid[cc];
    bool valid = (cur >= 0) && (pos >= d);
    row[d] = valid ? (float)cur : -1.0f;
    int cm = (cur - 1 < 0) ? 0 : (cur - 1);
    cur = (cur > 0) ? mask_index[cm] : -1;
  }
}

// ---------------------------------------------------------------------------
extern "C" void kernel_launch(void* const* d_in, const int* in_sizes, int n_in,
                              void* d_out, int out_size, void* d_ws, size_t ws_size,
                              hipStream_t stream) {
  const int*   mask_index = (const int*)d_in[0];
  const int*   tpid       = (const int*)d_in[1];
  const float* scores     = (const float*)d_in[2];

  const int T = in_sizes[0];         // 8191
  const int n = in_sizes[1];         // 8192
  const long long nn = (long long)n * n;
  const int max_depth = (int)(((long long)out_size - 2 * nn) / n);  // 16

  float* out_mask   = (float*)d_out;                       // n*n
  float* out_ri     = out_mask + nn;                       // n*max_depth
  float* out_masked = out_ri + (long long)n * max_depth;   // n*n

  int* flags = (int*)d_ws;     // n ints
  int* leaf  = flags + n;      // n ints

  // Big bandwidth-bound kernel first.
  rowfill_kernel<<<n, 256, 0, stream>>>(mask_index, scores, out_mask,
                                        out_masked, n, max_depth);

  // retrieve_indices pipeline (negligible cost).
  zero_flags_kernel<<<(n + 255) / 256, 256, 0, stream>>>(flags, n);
  mark_parents_kernel<<<(T + 255) / 256, 256, 0, stream>>>(mask_index, flags, T, n);
  leaf_scan_kernel<<<1, 1024, 0, stream>>>(flags, leaf, n);
  retrieve_kernel<<<(n + 255) / 256, 256, 0, stream>>>(leaf, mask_index, tpid,
                                                       out_ri, n, max_depth);
}